// GraphAttentionLayer_16458314678412
// MI455X (gfx1250) — compile-verified
//
#include <hip/hip_runtime.h>
#include <math.h>

// -------- problem constants (match reference) --------
#define IN_DIM      128
#define OUT_DIM     256
#define HEADS       4
#define NEG_SLOPE   0.2f
#define LN_EPS      1e-5f
#define SM_EPS      1e-16f

typedef __attribute__((ext_vector_type(2))) float v2f;
typedef __attribute__((ext_vector_type(8))) float v8f;

// ---------------------------------------------------------------------------
// Kernel 1: h = x @ W using V_WMMA_F32_16X16X4_F32.
// Block = 4 waves = one 16-row strip of x.
//   Stage 1: async-copy the 16x128 f32 A-strip (8 KB) global -> LDS using
//            global_load_async_to_lds_b128 (ASYNCcnt), s_wait_asynccnt,
//            workgroup barrier.
//   Stage 2: wave w computes output cols [w*64, w*64+64) as four 16x16
//            tiles, iterating K=128 in steps of 4; A-frags come from LDS
//            (ds_load_b64), B-frags from global (W stays L2/L0 hot).
// Register layouts (ISA 7.12.2):
//   A (16x4):  lane l -> M = l%16, K = kb + (l/16)*2 + {0,1}
//   B (4x16):  lane l -> N = l%16, K = kb + (l/16)*2 + {0,1}
//   D (16x16): lane l, vgpr r -> M = r + (l/16)*8, N = l%16
// ---------------------------------------------------------------------------
__global__ __launch_bounds__(128) void gat_gemm_wmma(
    const float* __restrict__ x, const float* __restrict__ W,
    float* __restrict__ hfeat, int Nnodes)
{
    __shared__ float atile[16 * IN_DIM];     // 8 KB, LDS offset 0

    const int tid  = threadIdx.x;
    const int wave = tid >> 5;
    const int lane = tid & 31;
    const int lm   = lane & 15;
    const int lh   = lane >> 4;              // 0 or 1 (upper lane half)
    const int m0   = blockIdx.x * 16;
    if (m0 >= Nnodes) return;                // uniform per block

    // ---- async stage A strip into LDS: 512 x 16B chunks, 4 per thread ----
    #pragma unroll
    for (int i = 0; i < 4; ++i) {
        const int chunk = tid + i * 128;                 // 0..511
        const int r     = chunk >> 5;                    // row 0..15
        const int c4    = chunk & 31;                    // 16B chunk in row
        const int grow  = min(m0 + r, Nnodes - 1);       // clamp tail strip
        const float* gp = x + (size_t)grow * IN_DIM + c4 * 4;
        const unsigned ldsoff =
            (unsigned)((r * IN_DIM + c4 * 4) * sizeof(float));
        asm volatile("global_load_async_to_lds_b128 %0, %1, off"
                     :: "v"(ldsoff), "v"(gp) : "memory");
    }
    asm volatile("s_wait_asynccnt 0" ::: "memory");
    __syncthreads();

    // ---- WMMA main loop ----
    const int nbase = wave * 64;
    v8f acc[4] = {};
    for (int k = 0; k < IN_DIM; k += 4) {
        v2f a = *(const v2f*)&atile[lm * IN_DIM + k + lh * 2];   // ds_load_b64
        #pragma unroll
        for (int t = 0; t < 4; ++t) {
            const int n = nbase + t * 16 + lm;
            v2f b;
            b.x = W[(size_t)(k + lh * 2 + 0) * OUT_DIM + n];
            b.y = W[(size_t)(k + lh * 2 + 1) * OUT_DIM + n];
            acc[t] = __builtin_amdgcn_wmma_f32_16x16x4_f32(
                false, a, false, b, (short)0, acc[t], false, false);
        }
    }

    // ---- store D tiles ----
    if (m0 + 16 <= Nnodes) {            // full strip: branch-free stores
        #pragma unroll
        for (int t = 0; t < 4; ++t) {
            float* hp = hfeat + (size_t)(m0 + lh * 8) * OUT_DIM
                              + nbase + t * 16 + lm;
            #pragma unroll
            for (int r = 0; r < 8; ++r)
                hp[(size_t)r * OUT_DIM] = acc[t][r];
        }
    } else {                            // tail strip (unused when N%16==0)
        #pragma unroll
        for (int t = 0; t < 4; ++t) {
            #pragma unroll
            for (int r = 0; r < 8; ++r) {
                const int m = m0 + r + lh * 8;
                if (m < Nnodes)
                    hfeat[(size_t)m * OUT_DIM + nbase + t * 16 + lm] = acc[t][r];
            }
        }
    }
}

// ---------------------------------------------------------------------------
// Kernel 2: a_src[n,h] = <h[n,h,:], att_src[h,:]>, same for a_dst.
// One wave per node; lane l handles flat channels [8l, 8l+8) (one head per
// 8-lane group), reduce within the 8-lane group via shfl_xor.
// ---------------------------------------------------------------------------
__global__ __launch_bounds__(256) void gat_attn_dots(
    const float* __restrict__ hfeat,
    const float* __restrict__ att_src, const float* __restrict__ att_dst,
    float* __restrict__ a_src, float* __restrict__ a_dst, int Nnodes)
{
    const int n = blockIdx.x * 8 + (threadIdx.x >> 5);
    if (n >= Nnodes) return;
    const int lane = threadIdx.x & 31;
    const float* hr = hfeat + (size_t)n * OUT_DIM + lane * 8;

    float ss = 0.f, sd = 0.f;
    #pragma unroll
    for (int j = 0; j < 8; ++j) {
        const float v = hr[j];
        ss += v * att_src[lane * 8 + j];   // att_* are [H,64] = flat 256, same
        sd += v * att_dst[lane * 8 + j];   // indexing as the h row
    }
    #pragma unroll
    for (int off = 1; off < 8; off <<= 1) {
        ss += __shfl_xor(ss, off, 32);
        sd += __shfl_xor(sd, off, 32);
    }
    if ((lane & 7) == 0) {
        const int head = lane >> 3;
        a_src[n * HEADS + head] = ss;
        a_dst[n * HEADS + head] = sd;
    }
}

// order-preserving float<->uint mapping for atomic max
__device__ __forceinline__ unsigned enc_ord(float f) {
    unsigned b = __float_as_uint(f);
    return (b & 0x80000000u) ? ~b : (b | 0x80000000u);
}
__device__ __forceinline__ float dec_ord(unsigned u) {
    unsigned b = (u & 0x80000000u) ? (u ^ 0x80000000u) : ~u;
    return __uint_as_float(b);
}

__device__ __forceinline__ void edge_nodes(const long long* __restrict__ ei,
                                           int e, int E, int& s, int& d) {
    if (e < E) { s = (int)ei[e]; d = (int)ei[(size_t)E + e]; }
    else       { s = d = e - E; }                 // appended self-loops
}

// ---------------------------------------------------------------------------
// Kernel 3: per-edge leaky-relu logit; segment max via ordered-uint atomicMax.
// Stores e to scratch so pass 4 need not recompute the gathers.
// ---------------------------------------------------------------------------
__global__ __launch_bounds__(256) void gat_edge_max(
    const long long* __restrict__ ei, int E, int ET,
    const float* __restrict__ a_src, const float* __restrict__ a_dst,
    float* __restrict__ e_store, unsigned* __restrict__ e_max_enc)
{
    const int e = blockIdx.x * blockDim.x + threadIdx.x;
    if (e >= ET) return;
    int s, d; edge_nodes(ei, e, E, s, d);
    #pragma unroll
    for (int hh = 0; hh < HEADS; ++hh) {
        float v = a_src[s * HEADS + hh] + a_dst[d * HEADS + hh];
        v = (v >= 0.f) ? v : NEG_SLOPE * v;
        e_store[(size_t)e * HEADS + hh] = v;
        atomicMax(&e_max_enc[d * HEADS + hh], enc_ord(v));
    }
}

// ---------------------------------------------------------------------------
// Kernel 4: e_exp = exp(e - max[dst]); denom[dst] += e_exp (in-place rewrite
// of e_store with the numerator).
// ---------------------------------------------------------------------------
__global__ __launch_bounds__(256) void gat_edge_exp(
    const long long* __restrict__ ei, int E, int ET,
    float* __restrict__ e_store, const unsigned* __restrict__ e_max_enc,
    float* __restrict__ denom)
{
    const int e = blockIdx.x * blockDim.x + threadIdx.x;
    if (e >= ET) return;
    int s, d; edge_nodes(ei, e, E, s, d);
    (void)s;
    #pragma unroll
    for (int hh = 0; hh < HEADS; ++hh) {
        const float m  = dec_ord(e_max_enc[d * HEADS + hh]);
        const float ex = expf(e_store[(size_t)e * HEADS + hh] - m);
        e_store[(size_t)e * HEADS + hh] = ex;
        atomicAdd(&denom[d * HEADS + hh], ex);
    }
}

// ---------------------------------------------------------------------------
// Kernel 5: out[dst] += alpha * h[src]. One wave per edge; lane l handles
// channels [8l,8l+8) (head = l/8). float4 gathers, scalar f32 atomic adds
// (global_atomic_add_f32). h/out are L2-resident (51 MB each, 192 MB L2).
// ---------------------------------------------------------------------------
__global__ __launch_bounds__(256) void gat_edge_scatter(
    const long long* __restrict__ ei, int E, int ET,
    const float* __restrict__ e_exp, const float* __restrict__ denom,
    const float* __restrict__ hfeat, float* __restrict__ out)
{
    const int e = blockIdx.x * 8 + (threadIdx.x >> 5);
    if (e >= ET) return;
    const int lane = threadIdx.x & 31;
    int s, d; edge_nodes(ei, e, E, s, d);

    const int head = lane >> 3;
    const float alpha = e_exp[(size_t)e * HEADS + head] /
                        (denom[d * HEADS + head] + SM_EPS);

    const float* hs = hfeat + (size_t)s * OUT_DIM + lane * 8;
    float*       op = out   + (size_t)d * OUT_DIM + lane * 8;
    const float4 v0 = *(const float4*)(hs);
    const float4 v1 = *(const float4*)(hs + 4);
    atomicAdd(op + 0, alpha * v0.x);
    atomicAdd(op + 1, alpha * v0.y);
    atomicAdd(op + 2, alpha * v0.z);
    atomicAdd(op + 3, alpha * v0.w);
    atomicAdd(op + 4, alpha * v1.x);
    atomicAdd(op + 5, alpha * v1.y);
    atomicAdd(op + 6, alpha * v1.z);
    atomicAdd(op + 7, alpha * v1.w);
}

// ---------------------------------------------------------------------------
// Kernel 6: out = relu(LN(out + bias) * gamma + beta). One wave per node.
// ---------------------------------------------------------------------------
__global__ __launch_bounds__(256) void gat_ln_relu(
    float* __restrict__ out, const float* __restrict__ bias,
    const float* __restrict__ gamma, const float* __restrict__ beta, int Nnodes)
{
    const int n = blockIdx.x * 8 + (threadIdx.x >> 5);
    if (n >= Nnodes) return;
    const int lane = threadIdx.x & 31;
    float* row = out + (size_t)n * OUT_DIM;

    float v[8];
    float sum = 0.f, sumsq = 0.f;
    #pragma unroll
    for (int j = 0; j < 8; ++j) {
        v[j] = row[lane * 8 + j] + bias[lane * 8 + j];
        sum   += v[j];
        sumsq += v[j] * v[j];
    }
    #pragma unroll
    for (int off = 1; off < 32; off <<= 1) {
        sum   += __shfl_xor(sum,   off, 32);
        sumsq += __shfl_xor(sumsq, off, 32);
    }
    const float mu  = sum * (1.f / OUT_DIM);
    const float var = sumsq * (1.f / OUT_DIM) - mu * mu;
    const float inv = rsqrtf(var + LN_EPS);
    #pragma unroll
    for (int j = 0; j < 8; ++j) {
        const float o = (v[j] - mu) * inv * gamma[lane * 8 + j] + beta[lane * 8 + j];
        row[lane * 8 + j] = fmaxf(o, 0.f);
    }
}

// ---------------------------------------------------------------------------
extern "C" void kernel_launch(void* const* d_in, const int* in_sizes, int n_in,
                              void* d_out, int out_size, void* d_ws, size_t ws_size,
                              hipStream_t stream)
{
    const float*     x       = (const float*)d_in[0];
    const long long* ei      = (const long long*)d_in[1];
    const float*     W       = (const float*)d_in[2];
    const float*     att_src = (const float*)d_in[3];
    const float*     att_dst = (const float*)d_in[4];
    const float*     bias    = (const float*)d_in[5];
    const float*     gamma   = (const float*)d_in[6];
    const float*     beta    = (const float*)d_in[7];
    float*           out     = (float*)d_out;

    const int Nn = in_sizes[0] / IN_DIM;     // 50000
    const int E  = in_sizes[1] / 2;          // 800000
    const int ET = E + Nn;                   // edges + self-loops

    // workspace layout
    char* ws = (char*)d_ws;
    float*    hfeat   = (float*)ws;    ws += (size_t)Nn * OUT_DIM * sizeof(float);
    float*    a_src   = (float*)ws;    ws += (size_t)Nn * HEADS  * sizeof(float);
    float*    a_dst   = (float*)ws;    ws += (size_t)Nn * HEADS  * sizeof(float);
    unsigned* e_max   = (unsigned*)ws; ws += (size_t)Nn * HEADS  * sizeof(unsigned);
    float*    denom   = (float*)ws;    ws += (size_t)Nn * HEADS  * sizeof(float);
    float*    e_store = (float*)ws;    ws += (size_t)ET * HEADS  * sizeof(float);

    hipMemsetAsync(e_max, 0, (size_t)Nn * HEADS * sizeof(unsigned), stream);
    hipMemsetAsync(denom, 0, (size_t)Nn * HEADS * sizeof(float), stream);
    hipMemsetAsync(out,   0, (size_t)out_size * sizeof(float), stream);

    gat_gemm_wmma   <<<(Nn + 15) / 16, 128, 0, stream>>>(x, W, hfeat, Nn);
    gat_attn_dots   <<<(Nn + 7) / 8,   256, 0, stream>>>(hfeat, att_src, att_dst,
                                                         a_src, a_dst, Nn);
    gat_edge_max    <<<(ET + 255)/256, 256, 0, stream>>>(ei, E, ET, a_src, a_dst,
                                                         e_store, e_max);
    gat_edge_exp    <<<(ET + 255)/256, 256, 0, stream>>>(ei, E, ET, e_store,
                                                         e_max, denom);
    gat_edge_scatter<<<(ET + 7) / 8,   256, 0, stream>>>(ei, E, ET, e_store,
                                                         denom, hfeat, out);
    gat_ln_relu     <<<(Nn + 7) / 8,   256, 0, stream>>>(out, bias, gamma, beta, Nn);
}